// Decoder_2284922602065
// MI455X (gfx1250) — compile-verified
//
#include <hip/hip_runtime.h>
#include <hip/hip_bf16.h>

// Problem constants (match reference)
constexpr int B_ = 32;
constexpr int W_ = 2048;
constexpr int H_ = 1024;
constexpr int E_ = 60;
constexpr int V_ = 100;
constexpr int KDEC = E_ + H_;     // 1084
constexpr int NPAD_OUT = 112;     // V padded to multiple of 16 for WMMA

typedef __attribute__((ext_vector_type(2))) float v2f;
typedef __attribute__((ext_vector_type(8))) float v8f;

// ---------------------------------------------------------------------------
// enc_len[b] = floor(src_len[b] * (W / src_len[0]) + 0.999)
__device__ __forceinline__ int compute_enc_len(const int* __restrict__ src_len, int b) {
    float scale = (float)W_ / (float)src_len[0];
    return (int)floorf((float)src_len[b] * scale + 0.999f);
}

__device__ __forceinline__ float sigmoidf_(float x) {
    return 1.0f / (1.0f + __expf(-x));
}

// ---------------------------------------------------------------------------
// h_mean = 0.5 * (hidden[0] + hidden[1])         (addMask = 1/L per layer)
__global__ __launch_bounds__(256)
void mean_kernel(const float* __restrict__ hidden, float* __restrict__ h_mean) {
    int idx = blockIdx.x * 256 + threadIdx.x;               // over B*H
    h_mean[idx] = 0.5f * (hidden[idx] + hidden[B_ * H_ + idx]);
}

// ---------------------------------------------------------------------------
// Generic skinny GEMM on the matrix pipes:
//   C[m,n] = sum_k X[m,k] * Wt[n,k] + bias[n]      (M = 32 rows)
// One wave per 16x16 output tile (4 waves/block), V_WMMA_F32_16X16X4_F32,
// K in steps of 4, two accumulator chains for ILP.
// A 16x4 layout: lanes 0-15 hold M=lane, VGPR{0,1}=K{0,1}; lanes 16-31: K{2,3}.
// B 4x16 layout mirrors A with N in place of M.
// C/D: VGPR r -> (M = r + 8*half, N = lane&15).
// MASKED=true handles an N tail (only the V=100 logits GEMM) so the five
// large weight-streaming GEMMs keep a pure load->wmma inner loop.
template <bool MASKED>
__global__ __launch_bounds__(128)
void gemm_xwt_wmma(const float* __restrict__ X, int ldx,
                   const float* __restrict__ Wt, int ldw,
                   const float* __restrict__ bias,
                   float* __restrict__ C, int ldc,
                   int Nreal, int K)
{
    const int lane = threadIdx.x & 31;
    const int wv   = threadIdx.x >> 5;          // wave id within block
    const int nwv  = blockDim.x >> 5;           // waves per block
    const int m0   = blockIdx.y * 16;
    const int n0   = (blockIdx.x * nwv + wv) * 16;
    const int half = lane >> 4;                 // 0 or 1
    const int lr   = lane & 15;
    const int kb   = half * 2;                  // this lane's K sub-offset

    const int arow = m0 + lr;
    int nrow = n0 + lr;
    float wmask = 1.0f;
    if (MASKED) {
        wmask = (nrow < Nreal) ? 1.0f : 0.0f;
        if (nrow >= Nreal) nrow = Nreal - 1;
    }

    const float* aptr = X  + (size_t)arow * ldx + kb;
    const float* bptr = Wt + (size_t)nrow * ldw + kb;

    v8f acc0 = {};
    v8f acc1 = {};
    int k = 0;
    for (; k + 8 <= K; k += 8) {
        v2f a0 = *(const v2f*)(aptr + k);
        v2f b0 = *(const v2f*)(bptr + k);
        v2f a1 = *(const v2f*)(aptr + k + 4);
        v2f b1 = *(const v2f*)(bptr + k + 4);
        if (MASKED) { b0 *= wmask; b1 *= wmask; }
        acc0 = __builtin_amdgcn_wmma_f32_16x16x4_f32(false, a0, false, b0,
                                                     (short)0, acc0, false, false);
        acc1 = __builtin_amdgcn_wmma_f32_16x16x4_f32(false, a1, false, b1,
                                                     (short)0, acc1, false, false);
    }
    for (; k < K; k += 4) {                     // K remainder (K=1084 case)
        v2f a0 = *(const v2f*)(aptr + k);
        v2f b0 = *(const v2f*)(bptr + k);
        if (MASKED) b0 *= wmask;
        acc0 = __builtin_amdgcn_wmma_f32_16x16x4_f32(false, a0, false, b0,
                                                     (short)0, acc0, false, false);
    }
    v8f acc = acc0 + acc1;

    const int col = n0 + lr;
    if (!MASKED || col < Nreal) {
        const float bv = bias[col];
#pragma unroll
        for (int r = 0; r < 8; ++r) {
            const int row = m0 + r + half * 8;
            C[(size_t)row * ldc + col] = acc[r] + bv;
        }
    }
}

// ---------------------------------------------------------------------------
// scores[b,w] = sum_h tanh(enc[w,b,h] + hidden_attn[b,h]) * W_oa[h] + b_oa
// One wave per (b,w); hidden_attn row + W_oa staged in LDS; only valid w.
__global__ __launch_bounds__(256)
void attn_scores_kernel(const float* __restrict__ enc,          // [W,B,H]
                        const float* __restrict__ hidden_attn,  // [B,H]
                        const float* __restrict__ W_oa,         // [H]
                        const float* __restrict__ b_oa,         // [1]
                        const int*   __restrict__ src_len,
                        float* __restrict__ scores)             // [B,W]
{
    __shared__ float s_ha[H_];
    __shared__ float s_w[H_];
    const int b   = blockIdx.y;
    const int tid = threadIdx.x;
    for (int i = tid; i < H_; i += 256) {
        s_ha[i] = hidden_attn[b * H_ + i];
        s_w[i]  = W_oa[i];
    }
    __syncthreads();

    const int enc_len = compute_enc_len(src_len, b);
    const int wave = tid >> 5;
    const int lane = tid & 31;
    const int w = blockIdx.x * 8 + wave;
    if (w >= enc_len) return;             // masked positions never read downstream

    const float* row = enc + ((size_t)w * B_ + b) * H_;
    float p = 0.0f;
    for (int k = lane; k < H_; k += 32)
        p += tanhf(row[k] + s_ha[k]) * s_w[k];
    for (int off = 16; off; off >>= 1)
        p += __shfl_xor(p, off, 32);
    if (lane == 0)
        scores[b * W_ + w] = p + b_oa[0];
}

// ---------------------------------------------------------------------------
// Masked softmax over w per batch; attn = 0 outside enc_len. One block per b.
__global__ __launch_bounds__(256)
void attn_softmax_kernel(const float* __restrict__ scores,
                         const int*   __restrict__ src_len,
                         float* __restrict__ attn)              // [B,W]
{
    __shared__ float red[256];
    const int b   = blockIdx.x;
    const int tid = threadIdx.x;
    const int enc_len = compute_enc_len(src_len, b);

    float mx = -3.4e38f;
    for (int w = tid; w < enc_len; w += 256)
        mx = fmaxf(mx, scores[b * W_ + w]);
    red[tid] = mx; __syncthreads();
    for (int s = 128; s; s >>= 1) {
        if (tid < s) red[tid] = fmaxf(red[tid], red[tid + s]);
        __syncthreads();
    }
    mx = red[0]; __syncthreads();

    float sum = 0.0f;
    for (int w = tid; w < enc_len; w += 256)
        sum += __expf(scores[b * W_ + w] - mx);
    red[tid] = sum; __syncthreads();
    for (int s = 128; s; s >>= 1) {
        if (tid < s) red[tid] += red[tid + s];
        __syncthreads();
    }
    const float inv = 1.0f / red[0];

    for (int w = tid; w < W_; w += 256) {
        float a = 0.0f;
        if (w < enc_len)
            a = __expf(scores[b * W_ + w] - mx) * inv;
        attn[b * W_ + w] = a;
    }
}

// ---------------------------------------------------------------------------
// context[b,h] = sum_{w < enc_len[b]} attn[b,w] * enc[w,b,h]
// Block = (hchunk=256, b); attn chunk staged in LDS; prefetch strided rows.
__global__ __launch_bounds__(256)
void context_kernel(const float* __restrict__ enc,
                    const float* __restrict__ attn,
                    const int*   __restrict__ src_len,
                    float* __restrict__ context)                // [B,H]
{
    __shared__ float s_a[256];
    const int b   = blockIdx.y;
    const int h   = blockIdx.x * 256 + threadIdx.x;
    const int enc_len = compute_enc_len(src_len, b);

    float acc = 0.0f;
    for (int w0 = 0; w0 < enc_len; w0 += 256) {
        __syncthreads();
        const int widx = w0 + threadIdx.x;
        s_a[threadIdx.x] = (widx < enc_len) ? attn[b * W_ + widx] : 0.0f;
        __syncthreads();
        const int wmax = min(256, enc_len - w0);
        for (int j = 0; j < wmax; ++j) {
            const float* p = enc + ((size_t)(w0 + j) * B_ + b) * H_ + h;
            if (j + 1 < wmax)
                __builtin_prefetch(p + (size_t)B_ * H_, 0, 0);  // next 128KB-strided row
            acc = fmaf(s_a[j], *p, acc);
        }
    }
    context[b * H_ + h] = acc;
}

// ---------------------------------------------------------------------------
// top1 = argmax(in_char[b]); in_dec[b] = [embedding[top1] | context[b]]
__global__ __launch_bounds__(32)
void embed_concat_kernel(const float* __restrict__ in_char,     // [B,V]
                         const float* __restrict__ embedding,   // [V,E]
                         const float* __restrict__ context,     // [B,H]
                         float* __restrict__ in_dec)            // [B,KDEC]
{
    const int b = blockIdx.x, lane = threadIdx.x;
    float best = -3.4e38f;
    int   bi   = 0x7fffffff;
    for (int v = lane; v < V_; v += 32) {
        float x = in_char[b * V_ + v];
        if (x > best) { best = x; bi = v; }
    }
    for (int off = 16; off; off >>= 1) {
        float ob = __shfl_xor(best, off, 32);
        int   oi = __shfl_xor(bi, off, 32);
        if (ob > best || (ob == best && oi < bi)) { best = ob; bi = oi; }
    }
    const int top1 = __shfl(bi, 0, 32);
    for (int e = lane; e < E_; e += 32)
        in_dec[(size_t)b * KDEC + e] = embedding[top1 * E_ + e];
    for (int h = lane; h < H_; h += 32)
        in_dec[(size_t)b * KDEC + E_ + h] = context[b * H_ + h];
}

// ---------------------------------------------------------------------------
// GRU gate math (biases already folded into gi/gh by the GEMM):
//   r = sig(ir+hr); z = sig(iz+hz); n = tanh(inn + r*hn); h' = (1-z)n + z h
__global__ __launch_bounds__(256)
void gru_gate_kernel(const float* __restrict__ gi,     // [B,3H]
                     const float* __restrict__ gh,     // [B,3H]
                     const float* __restrict__ hprev,  // [B,H]
                     float* __restrict__ hnew)         // [B,H]
{
    const int idx = blockIdx.x * 256 + threadIdx.x;    // over B*H
    const int b = idx / H_, j = idx - b * H_;
    const float* gib = gi + (size_t)b * 3 * H_;
    const float* ghb = gh + (size_t)b * 3 * H_;
    const float ir = gib[j],          hr = ghb[j];
    const float iz = gib[H_ + j],     hz = ghb[H_ + j];
    const float in = gib[2 * H_ + j], hn = ghb[2 * H_ + j];
    const float r = sigmoidf_(ir + hr);
    const float z = sigmoidf_(iz + hz);
    const float n = tanhf(in + r * hn);
    hnew[idx] = (1.0f - z) * n + z * hprev[idx];
}

// ---------------------------------------------------------------------------
// output[b, :] = softmax(logits[b, :V])  (logits padded to NPAD_OUT columns)
__global__ __launch_bounds__(128)
void out_softmax_kernel(const float* __restrict__ logits,       // [B,NPAD_OUT]
                        float* __restrict__ out)                // [B,V]
{
    __shared__ float red[128];
    const int b = blockIdx.x, tid = threadIdx.x;
    const float x = (tid < V_) ? logits[b * NPAD_OUT + tid] : -3.4e38f;
    red[tid] = x; __syncthreads();
    for (int s = 64; s; s >>= 1) {
        if (tid < s) red[tid] = fmaxf(red[tid], red[tid + s]);
        __syncthreads();
    }
    const float mx = red[0]; __syncthreads();
    const float e = (tid < V_) ? __expf(x - mx) : 0.0f;
    red[tid] = e; __syncthreads();
    for (int s = 64; s; s >>= 1) {
        if (tid < s) red[tid] += red[tid + s];
        __syncthreads();
    }
    if (tid < V_) out[b * V_ + tid] = e / red[0];
}

// ---------------------------------------------------------------------------
extern "C" void kernel_launch(void* const* d_in, const int* in_sizes, int n_in,
                              void* d_out, int out_size, void* d_ws, size_t ws_size,
                              hipStream_t stream) {
    const float* in_char   = (const float*)d_in[0];
    const float* hidden    = (const float*)d_in[1];   // [2,B,H]
    const float* enc       = (const float*)d_in[2];   // [W,B,H]
    const int*   src_len   = (const int*)  d_in[3];
    const float* embedding = (const float*)d_in[4];
    const float* W_hp      = (const float*)d_in[5];
    const float* b_hp      = (const float*)d_in[6];
    const float* W_oa      = (const float*)d_in[7];
    const float* b_oa      = (const float*)d_in[8];
    const float* Wih0      = (const float*)d_in[9];
    const float* Whh0      = (const float*)d_in[10];
    const float* bih0      = (const float*)d_in[11];
    const float* bhh0      = (const float*)d_in[12];
    const float* Wih1      = (const float*)d_in[13];
    const float* Whh1      = (const float*)d_in[14];
    const float* bih1      = (const float*)d_in[15];
    const float* bhh1      = (const float*)d_in[16];
    const float* W_out     = (const float*)d_in[17];
    const float* b_out     = (const float*)d_in[18];

    // d_out layout (return order, flat): output | latest_hidden | attn
    float* out_probs = (float*)d_out;                         // [B,V]
    float* h0_out    = out_probs + B_ * V_;                   // [B,H]
    float* h1_out    = h0_out + B_ * H_;                      // [B,H]
    float* attn      = h1_out + B_ * H_;                      // [B,W]

    // workspace layout (floats)
    float* ws          = (float*)d_ws;
    float* h_mean      = ws;                                  // 32768
    float* hidden_attn = h_mean + B_ * H_;                    // 32768
    float* scores      = hidden_attn + B_ * H_;               // 65536
    float* context     = scores + B_ * W_;                    // 32768
    float* in_dec      = context + B_ * H_;                   // 34688
    float* gi          = in_dec + B_ * KDEC;                  // 98304
    float* gh          = gi + B_ * 3 * H_;                    // 98304
    float* logits      = gh + B_ * 3 * H_;                    // 3584

    const float* hidden0 = hidden;
    const float* hidden1 = hidden + B_ * H_;

    // 1) h_mean
    mean_kernel<<<(B_ * H_) / 256, 256, 0, stream>>>(hidden, h_mean);

    // 2) hidden_attn = h_mean @ W_hp^T + b_hp   [32,1024]x[1024,1024]
    gemm_xwt_wmma<false><<<dim3(H_ / 16 / 4, B_ / 16), 128, 0, stream>>>(
        h_mean, H_, W_hp, H_, b_hp, hidden_attn, H_, H_, H_);

    // 3) attention scores (only valid w)
    attn_scores_kernel<<<dim3(W_ / 8, B_), 256, 0, stream>>>(
        enc, hidden_attn, W_oa, b_oa, src_len, scores);

    // 4) masked softmax -> attn (directly into d_out)
    attn_softmax_kernel<<<B_, 256, 0, stream>>>(scores, src_len, attn);

    // 5) context = attn-weighted sum of encoder rows
    context_kernel<<<dim3(H_ / 256, B_), 256, 0, stream>>>(enc, attn, src_len, context);

    // 6) argmax + embed + concat
    embed_concat_kernel<<<B_, 32, 0, stream>>>(in_char, embedding, context, in_dec);

    // 7) GRU layer 0: gi = in_dec @ Wih0^T + bih0 ; gh = h_prev0 @ Whh0^T + bhh0
    gemm_xwt_wmma<false><<<dim3((3 * H_) / 16 / 4, B_ / 16), 128, 0, stream>>>(
        in_dec, KDEC, Wih0, KDEC, bih0, gi, 3 * H_, 3 * H_, KDEC);
    gemm_xwt_wmma<false><<<dim3((3 * H_) / 16 / 4, B_ / 16), 128, 0, stream>>>(
        hidden0, H_, Whh0, H_, bhh0, gh, 3 * H_, 3 * H_, H_);
    gru_gate_kernel<<<(B_ * H_) / 256, 256, 0, stream>>>(gi, gh, hidden0, h0_out);

    // 8) GRU layer 1
    gemm_xwt_wmma<false><<<dim3((3 * H_) / 16 / 4, B_ / 16), 128, 0, stream>>>(
        h0_out, H_, Wih1, H_, bih1, gi, 3 * H_, 3 * H_, H_);
    gemm_xwt_wmma<false><<<dim3((3 * H_) / 16 / 4, B_ / 16), 128, 0, stream>>>(
        hidden1, H_, Whh1, H_, bhh1, gh, 3 * H_, 3 * H_, H_);
    gru_gate_kernel<<<(B_ * H_) / 256, 256, 0, stream>>>(gi, gh, hidden1, h1_out);

    // 9) logits = h1 @ W_out^T + b_out   (N padded 112, real 100), then softmax
    gemm_xwt_wmma<true><<<dim3(NPAD_OUT / 16, B_ / 16), 32, 0, stream>>>(
        h1_out, H_, W_out, H_, b_out, logits, NPAD_OUT, V_, H_);
    out_softmax_kernel<<<B_, 128, 0, stream>>>(logits, out_probs);
}